// MeshConvPoint_74208444940566
// MI455X (gfx1250) — compile-verified
//
#include <hip/hip_runtime.h>

typedef float v2f __attribute__((ext_vector_type(2)));
typedef float v8f __attribute__((ext_vector_type(8)));

#define BB 8
#define CC 64
#define VV 25000
#define DD 12
#define OO 64

// ---------------------------------------------------------------------------
// Transpose x[B,C,V] -> xT[B,V,C] so neighbor gathers become contiguous 256B
// rows (coalesced global_load_b128 across a half-wave).
// ---------------------------------------------------------------------------
__global__ __launch_bounds__(256) void transpose_kernel(const float* __restrict__ x,
                                                        float* __restrict__ xT) {
  __shared__ float tile[CC][33];   // pad to avoid bank conflicts
  const int b  = blockIdx.y;
  const int v0 = blockIdx.x * 32;
  const int t  = threadIdx.x;

  const int vr = t & 31;   // read-phase v (coalesced along V)
  const int cr = t >> 5;   // 0..7
#pragma unroll
  for (int r = 0; r < 8; ++r) {
    int c = r * 8 + cr;
    int v = v0 + vr;
    if (v < VV) tile[c][vr] = x[((size_t)b * CC + c) * VV + v];
  }
  __syncthreads();

  const int cw = t & 63;   // write-phase c (coalesced along C)
  const int vw = t >> 6;   // 0..3
#pragma unroll
  for (int r = 0; r < 8; ++r) {
    int vl = r * 4 + vw;
    int v  = v0 + vl;
    if (v < VV) xT[((size_t)b * VV + v) * CC + cw] = tile[cw][vl];
  }
}

// ---------------------------------------------------------------------------
// Fused gather-mean + pointwise conv via fp32 WMMA.
// One 128-thread workgroup handles (batch b, 64-point tile).
// B-tile in LDS: Bmat[point][k], k in [0,64) = self feature, [64,128) = mean.
// A (weights) in LDS transposed: Wl[k][o].
// Each wave computes Out[64, 16] = A[64,128] x B[128,16] + bias with
// 4 M-tiles x 32 chained v_wmma_f32_16x16x4_f32.
// ---------------------------------------------------------------------------
__global__ __launch_bounds__(128) void meshconv_kernel(
    const float* __restrict__ xg,   // xT [B,V,C] if useT else x [B,C,V]
    const int*   __restrict__ nbr,  // [B,V,D]
    const int*   __restrict__ deg,  // [B,V]
    const float* __restrict__ W,    // [O,C,2]
    const float* __restrict__ bias, // [O]
    float* __restrict__ out,        // [B,O,V]
    int useT)
{
  __shared__ __align__(16) float Bmat[64][132]; // pitch 132: conflict-free frag reads
  __shared__ float Wl[128][17];                 // pitch 17:  conflict-free frag reads

  const int b    = blockIdx.y;
  const int v0   = blockIdx.x * 64;
  const int t    = threadIdx.x;
  const int wave = t >> 5;
  const int lane = t & 31;
  const int half = lane >> 4;     // 0: lanes 0-15, 1: lanes 16-31
  const int li   = lane & 15;
  const int c0   = li * 4;        // 4 channels per lane in gather phase

  // ---- Stage Wflat[o][k] transposed into LDS: Wl[k][o] ----
  // Wflat[o][k] = W[o, k%64, k/64]  (k<64: self weight, k>=64: mean weight)
  for (int i = t; i < 128 * 64; i += 128) {
    int k = i >> 6, o = i & 63;
    Wl[k][o] = W[o * 128 + ((k & 63) << 1) + (k >> 6)];
  }

  // ---- Gather + masked mean: half-wave per point, 4 channels per lane ----
  for (int pp = 0; pp < 8; ++pp) {
    int pl = wave * 16 + pp * 2 + half;          // local point 0..63
    int v  = v0 + pl;
    int vc = v < VV ? v : VV - 1;                // clamp edge tile (store is guarded)
    int base = b * VV + vc;
    int dg = deg[base];

    float4 m = make_float4(0.f, 0.f, 0.f, 0.f);
    for (int d = 0; d < DD; ++d) {
      if (d < dg) {
        int ni = nbr[base * DD + d];
        float4 tt;
        if (useT) {
          tt = *(const float4*)(xg + ((size_t)b * VV + ni) * CC + c0); // coalesced
        } else {
          const float* xp = xg + ((size_t)b * CC + c0) * VV + ni;     // strided fallback
          tt.x = xp[0]; tt.y = xp[VV]; tt.z = xp[2 * VV]; tt.w = xp[3 * VV];
        }
        m.x += tt.x; m.y += tt.y; m.z += tt.z; m.w += tt.w;
      }
    }
    float inv = 1.0f / (float)(dg > 0 ? dg : 1);

    float4 s;
    if (useT) {
      s = *(const float4*)(xg + ((size_t)b * VV + vc) * CC + c0);
    } else {
      const float* xp = xg + ((size_t)b * CC + c0) * VV + vc;
      s.x = xp[0]; s.y = xp[VV]; s.z = xp[2 * VV]; s.w = xp[3 * VV];
    }

    *(float4*)&Bmat[pl][c0]      = s;                                       // k = c
    *(float4*)&Bmat[pl][64 + c0] = make_float4(m.x*inv, m.y*inv, m.z*inv, m.w*inv); // k = 64+c
  }
  __syncthreads();

  // ---- Hoist this wave's 32 B fragments (K=4 each) into VGPRs ----
  // B 4x16 layout: VGPR0 = rows K+0 | K+2 (lane halves), VGPR1 = rows K+1 | K+3.
  float bf0[32], bf1[32];
  const int n  = wave * 16 + li;   // column (point) this lane owns
  const int kb = half * 2;         // K sub-offset for this half-wave
#pragma unroll
  for (int kk = 0; kk < 32; ++kk) {
    bf0[kk] = Bmat[n][kk * 4 + kb];
    bf1[kk] = Bmat[n][kk * 4 + kb + 1];
  }

  const int vv = v0 + wave * 16 + li;
#pragma unroll
  for (int mt = 0; mt < 4; ++mt) {
    // C init with bias: VGPR g holds rows M=g (lanes 0-15) / M=g+8 (lanes 16-31)
    v8f acc;
#pragma unroll
    for (int g = 0; g < 8; ++g) acc[g] = bias[mt * 16 + g + half * 8];

#pragma unroll
    for (int kk = 0; kk < 32; ++kk) {
      int Ka = kk * 4 + kb;
      v2f a, bv;
      // A 16x4 layout: lane holds row M=lane%16; VGPR0 = K+0|K+2, VGPR1 = K+1|K+3
      a.x = Wl[Ka][mt * 16 + li];
      a.y = Wl[Ka + 1][mt * 16 + li];
      bv.x = bf0[kk];
      bv.y = bf1[kk];
      acc = __builtin_amdgcn_wmma_f32_16x16x4_f32(
          /*neg_a=*/false, a, /*neg_b=*/false, bv,
          /*c_mod=*/(short)0, acc, /*reuse_a=*/false, /*reuse_b=*/false);
    }

    if (vv < VV) {
#pragma unroll
      for (int g = 0; g < 8; ++g) {
        int o = mt * 16 + g + half * 8;
        out[((size_t)b * OO + o) * VV + vv] = acc[g];  // coalesced 64B rows
      }
    }
  }
}

// ---------------------------------------------------------------------------
extern "C" void kernel_launch(void* const* d_in, const int* in_sizes, int n_in,
                              void* d_out, int out_size, void* d_ws, size_t ws_size,
                              hipStream_t stream) {
  const float* x    = (const float*)d_in[0];
  const int*   nbr  = (const int*)d_in[1];
  const int*   deg  = (const int*)d_in[2];
  const float* W    = (const float*)d_in[3];
  const float* bias = (const float*)d_in[4];
  float* out = (float*)d_out;

  const size_t xT_bytes = (size_t)BB * VV * CC * sizeof(float);
  const int useT = (d_ws != nullptr && ws_size >= xT_bytes) ? 1 : 0;

  const float* xsrc = x;
  if (useT) {
    float* xT = (float*)d_ws;
    transpose_kernel<<<dim3((VV + 31) / 32, BB), 256, 0, stream>>>(x, xT);
    xsrc = xT;
  }
  meshconv_kernel<<<dim3((VV + 63) / 64, BB), 128, 0, stream>>>(
      xsrc, nbr, deg, W, bias, out, useT);
}